// SelfAttentiveBimodalFusion_PTBlock_51393578664166
// MI455X (gfx1250) — compile-verified
//
#include <hip/hip_runtime.h>

#define N 32768
#define NS 16
#define C_MAIN 128
#define C_FUSED 256
#define C_IN 16
#define KNN_CHUNK 2048

typedef __attribute__((ext_vector_type(2))) float v2f;
typedef __attribute__((ext_vector_type(8))) float v8f;

__device__ __forceinline__ v8f wmma_f32_16x16x4(v2f a, v2f b, v8f c) {
  // 8 args: (neg_a, A, neg_b, B, c_mod, C, reuse_a, reuse_b)
  return __builtin_amdgcn_wmma_f32_16x16x4_f32(false, a, false, b, (short)0, c,
                                               false, false);
}

// ---------------------------------------------------------------------------
// Kernel 1: exact kNN (k=16) via WMMA f32 16x16x4 distance tiles.
// A = candidates (px,py,pz,|p|^2), B = queries (-2qx,-2qy,-2qz,1)
//   =>  D[m][n] = |p_m|^2 - 2 p_m.q_n   (ranks identically to squared dist).
// Pipeline per 2048-candidate chunk:
//   1) global_load_async_to_lds_b128 raw xyz -> aux   (ASYNCcnt path)
//   2) one cooperative pass: aux -> cb4[(x,y,z,|p|^2)] (norm computed once
//      per block, not once per wave per tile)
//   3) scan: A fragment is a single ds_load_b64 (lo lanes: x,y; hi lanes:
//      z,|p|^2), 1 WMMA + min3-tree early-out per 16 candidates; branchless
//      cndmask top-16 insertion. Tile loop unrolled x2 for two independent
//      WMMA chains.
// Lanes l and l+16 own query (l&15) over disjoint candidate halves; exact
// 2-list merge at the end reuses the staging LDS (aux) for the lists.
// ---------------------------------------------------------------------------
__global__ __launch_bounds__(256) void knn_kernel(const float* __restrict__ xyz,
                                                  int* __restrict__ idx_out) {
  __shared__ float cb4[KNN_CHUNK * 4];  // (x,y,z,|p|^2) per candidate (32KB)
  __shared__ float aux[8192];           // staging (24KB) / merge lists (32KB)

  const int tid  = threadIdx.x;
  const int wave = tid >> 5;
  const int lane = tid & 31;
  const int hi   = lane >> 4;
  const int lo   = lane & 15;
  const int qbase = (blockIdx.x * 8 + wave) * 16;

  // B fragment (query tile): lanes 0-15 hold K rows {0,1}, lanes 16-31 {2,3}
  const int qn = qbase + lo;
  const float qx = xyz[qn * 3 + 0];
  const float qy = xyz[qn * 3 + 1];
  const float qz = xyz[qn * 3 + 2];
  v2f bfrag;
  bfrag[0] = hi ? (-2.0f * qz) : (-2.0f * qx);
  bfrag[1] = hi ? 1.0f : (-2.0f * qy);

  float bd[16];
  int   bi[16];
#pragma unroll
  for (int k = 0; k < 16; ++k) { bd[k] = 3.0e38f; bi[k] = 0; }

  const unsigned stage_lds = (unsigned)(uintptr_t)(&aux[0]);
  // per-lane A-fragment source: candidate m -> cb4 + m*4 + hi*2 (b64 load)
  const float* afp = cb4 + lo * 4 + hi * 2;

  // one 16-candidate tile: ds_load_b64 -> WMMA -> min3 early-out -> insert
  auto process_tile = [&](int chunk, int ct) {
    const v2f a = *(const v2f*)(afp + ct * 4);
    v8f acc = {0.f, 0.f, 0.f, 0.f, 0.f, 0.f, 0.f, 0.f};
    acc = wmma_f32_16x16x4(a, bfrag, acc);

    const float tmin =
        fminf(fminf(fminf(acc[0], acc[1]), fminf(acc[2], acc[3])),
              fminf(fminf(acc[4], acc[5]), fminf(acc[6], acc[7])));
    if (tmin < bd[15]) {
      const int jbase = chunk + ct + 8 * hi;
#pragma unroll
      for (int r = 0; r < 8; ++r) {
        const float s = acc[r];
        if (s < bd[15]) {
          float cs = s;
          int   ci = jbase + r;
          // branchless sorted insertion: cmp + 4 cndmask per level
#pragma unroll
          for (int k = 0; k < 16; ++k) {
            const float ob = bd[k];
            const int   oi = bi[k];
            const bool  lt = cs < ob;
            bd[k] = lt ? cs : ob;
            bi[k] = lt ? ci : oi;
            cs    = lt ? ob : cs;
            ci    = lt ? oi : ci;
          }
        }
      }
    }
  };

  for (int chunk = 0; chunk < N; chunk += KNN_CHUNK) {
    __syncthreads();  // previous chunk fully consumed before overwrite
    // (1) async stage: 2048 * 12B = 24576B = 1536 x B128; 6 ops per thread
#pragma unroll
    for (int t = 0; t < (KNN_CHUNK * 12) / (16 * 256); ++t) {
      const int op = t * 256 + tid;
      const unsigned long long gaddr =
          (unsigned long long)(uintptr_t)(xyz) +
          (unsigned long long)(chunk * 12 + op * 16);
      const unsigned dst = stage_lds + op * 16;
      asm volatile("global_load_async_to_lds_b128 %0, %1, off"
                   :: "v"(dst), "v"(gaddr) : "memory");
    }
    asm volatile("s_wait_asynccnt 0x0" ::: "memory");
    __syncthreads();

    // (2) transform: packed float3 -> (x,y,z,|p|^2) float4 table
#pragma unroll
    for (int t = 0; t < KNN_CHUNK / 256; ++t) {
      const int c = t * 256 + tid;
      const float x = aux[c * 3 + 0];
      const float y = aux[c * 3 + 1];
      const float z = aux[c * 3 + 2];
      *(float4*)&cb4[c * 4] = make_float4(x, y, z, fmaf(x, x, fmaf(y, y, z * z)));
    }
    __syncthreads();

    // (3) scan: 128 tiles, unrolled x2 for independent WMMA chains
    for (int ct = 0; ct < KNN_CHUNK; ct += 32) {
      process_tile(chunk, ct);
      process_tile(chunk, ct + 16);
    }
  }

  // exact merge of the two half-lists per query (lane q and lane q+16);
  // reuse aux (staging dead since last transform barrier) for the lists.
  float* md = aux;                       // 4096 floats
  int*   mi = (int*)(aux + 4096);        // 4096 ints
#pragma unroll
  for (int k = 0; k < 16; ++k) {
    md[(wave * 32 + lane) * 16 + k] = bd[k];
    mi[(wave * 32 + lane) * 16 + k] = bi[k];
  }
  __syncthreads();
  if (lane < 16) {
    const float* da = &md[(wave * 32 + lane) * 16];
    const int*   ia = &mi[(wave * 32 + lane) * 16];
    const float* db = &md[(wave * 32 + lane + 16) * 16];
    const int*   ib = &mi[(wave * 32 + lane + 16) * 16];
    int pa = 0, pb = 0;
    for (int t = 0; t < NS; ++t) {
      const float va = da[pa], vb = db[pb];
      int sel;
      if (va <= vb) { sel = ia[pa]; ++pa; }
      else          { sel = ib[pb]; ++pb; }
      idx_out[(qbase + lane) * NS + t] = sel;      // ascending-distance order
    }
  }
}

// ---------------------------------------------------------------------------
// Kernel 2: h = concat(x_main,x_mod) @ W_ein + b_ein (K=256 via 64 WMMAs),
// then q/k/v = h @ W{q,k,v} + b (K=16 via 4 WMMAs each). h-tile round-trips
// through LDS to convert D-layout -> A-layout.
// ---------------------------------------------------------------------------
__global__ __launch_bounds__(256) void ein_qkv_kernel(
    const float* __restrict__ x_main, const float* __restrict__ x_mod,
    const float* __restrict__ W_ein, const float* __restrict__ b_ein,
    const float* __restrict__ Wq, const float* __restrict__ bq,
    const float* __restrict__ Wk, const float* __restrict__ bk,
    const float* __restrict__ Wv, const float* __restrict__ bv,
    float* __restrict__ xq, float* __restrict__ xk, float* __restrict__ xv) {
  __shared__ float sWT[16 * 256];       // sWT[c*256+k] = W_ein[k*16+c]
  __shared__ float sh[8][16][17];       // per-wave h tile (padded)

  const int tid  = threadIdx.x;
  const int wave = tid >> 5;
  const int lane = tid & 31;
  const int hi   = lane >> 4;
  const int lo   = lane & 15;
  const int rb   = (blockIdx.x * 8 + wave) * 16;

#pragma unroll
  for (int i = 0; i < 16; ++i) {
    const int e = tid * 16 + i;         // 4096 weights
    sWT[(e & 15) * 256 + (e >> 4)] = W_ein[e];
  }
  __syncthreads();

  const int row = rb + lo;              // A row owned by this lane
  const float* xm = x_main + row * C_MAIN;
  const float* xo = x_mod  + row * C_MAIN;

  v8f acc = {0.f, 0.f, 0.f, 0.f, 0.f, 0.f, 0.f, 0.f};
#pragma unroll 4
  for (int kk = 0; kk < C_FUSED; kk += 4) {
    const int k0 = kk + 2 * hi;         // this lane's K pair
    const float* src = (k0 < C_MAIN) ? (xm + k0) : (xo + (k0 - C_MAIN));
    v2f a; a[0] = src[0]; a[1] = src[1];
    v2f b; b[0] = sWT[lo * 256 + k0]; b[1] = sWT[lo * 256 + k0 + 1];
    acc = wmma_f32_16x16x4(a, b, acc);
  }

  const float be = b_ein[lo];
#pragma unroll
  for (int r = 0; r < 8; ++r) sh[wave][r + 8 * hi][lo] = acc[r] + be;
  __syncthreads();

  v2f ha[4];
#pragma unroll
  for (int t = 0; t < 4; ++t) {
    const int k0 = 4 * t + 2 * hi;
    ha[t][0] = sh[wave][lo][k0];
    ha[t][1] = sh[wave][lo][k0 + 1];
  }

  v8f qa = {0.f, 0.f, 0.f, 0.f, 0.f, 0.f, 0.f, 0.f};
  v8f ka = qa, va = qa;
#pragma unroll
  for (int t = 0; t < 4; ++t) {
    const int k0 = 4 * t + 2 * hi;
    v2f b;
    b[0] = Wq[k0 * 16 + lo]; b[1] = Wq[(k0 + 1) * 16 + lo];
    qa = wmma_f32_16x16x4(ha[t], b, qa);
    b[0] = Wk[k0 * 16 + lo]; b[1] = Wk[(k0 + 1) * 16 + lo];
    ka = wmma_f32_16x16x4(ha[t], b, ka);
    b[0] = Wv[k0 * 16 + lo]; b[1] = Wv[(k0 + 1) * 16 + lo];
    va = wmma_f32_16x16x4(ha[t], b, va);
  }
  const float cbq = bq[lo], cbk = bk[lo], cbv = bv[lo];
#pragma unroll
  for (int r = 0; r < 8; ++r) {
    const int orow = rb + r + 8 * hi;
    xq[orow * 16 + lo] = qa[r] + cbq;
    xk[orow * 16 + lo] = ka[r] + cbk;
    xv[orow * 16 + lo] = va[r] + cbv;
  }
}

// ---------------------------------------------------------------------------
// Kernel 3: per-point local vector attention (16 neighbors). One thread per
// point; BN folded to scale/shift; two passes over neighbors (recompute p_e
// on pass 2 instead of holding 256 live floats). Raw logits kept in padded
// LDS (bank-conflict-free stride 33).
// ---------------------------------------------------------------------------
__global__ __launch_bounds__(256) void attn_kernel(
    const float* __restrict__ xyz, const int* __restrict__ idx,
    const float* __restrict__ xq, const float* __restrict__ xk,
    const float* __restrict__ xv,
    const float* __restrict__ Wp1, const float* __restrict__ bp1,
    const float* __restrict__ Wp2, const float* __restrict__ bp2,
    const float* __restrict__ Wl1, const float* __restrict__ bl1,
    const float* __restrict__ Wl2, const float* __restrict__ bl2,
    const float* __restrict__ bnp_g, const float* __restrict__ bnp_b,
    const float* __restrict__ bnp_m, const float* __restrict__ bnp_v,
    const float* __restrict__ bw1_g, const float* __restrict__ bw1_b,
    const float* __restrict__ bw1_m, const float* __restrict__ bw1_v,
    const float* __restrict__ bw2_g, const float* __restrict__ bw2_b,
    const float* __restrict__ bw2_m, const float* __restrict__ bw2_v,
    float* __restrict__ agg_out) {
  __shared__ float sWp1[9], sbp1v[3], sps[3], spt[3];
  __shared__ float sWp2[48], sbp2[16];
  __shared__ float s1[16], t1[16];
  __shared__ float sWl1[32], sbl1[2], s2c[2], t2c[2], sWl2[4], sbl2[2];
  __shared__ float wvs[256][33];        // 16 neighbors x 2 logits, padded

  const int tid = threadIdx.x;
  if (tid < 9)  sWp1[tid] = Wp1[tid];
  if (tid < 48) sWp2[tid] = Wp2[tid];
  if (tid < 32) sWl1[tid] = Wl1[tid];
  if (tid < 4)  sWl2[tid] = Wl2[tid];
  if (tid < 3) {
    sbp1v[tid] = bp1[tid];
    const float s = bnp_g[tid] * rsqrtf(bnp_v[tid] + 1e-5f);
    sps[tid] = s; spt[tid] = bnp_b[tid] - bnp_m[tid] * s;
  }
  if (tid < 16) {
    sbp2[tid] = bp2[tid];
    const float s = bw1_g[tid] * rsqrtf(bw1_v[tid] + 1e-5f);
    s1[tid] = s; t1[tid] = bw1_b[tid] - bw1_m[tid] * s;
  }
  if (tid < 2) {
    sbl1[tid] = bl1[tid]; sbl2[tid] = bl2[tid];
    const float s = bw2_g[tid] * rsqrtf(bw2_v[tid] + 1e-5f);
    s2c[tid] = s; t2c[tid] = bw2_b[tid] - bw2_m[tid] * s;
  }
  __syncthreads();

  const int i = blockIdx.x * 256 + tid;
  const float qx = xyz[i * 3 + 0], qy = xyz[i * 3 + 1], qz = xyz[i * 3 + 2];
  float q[16];
#pragma unroll
  for (int c = 0; c < 16; ++c) q[c] = xq[i * 16 + c];
  const int* nb = idx + i * NS;

  // pass 1: logits
#pragma unroll 1
  for (int n = 0; n < NS; ++n) {
    const int j = nb[n];
    const float dx = xyz[j * 3 + 0] - qx;
    const float dy = xyz[j * 3 + 1] - qy;
    const float dz = xyz[j * 3 + 2] - qz;
    float tv[3];
#pragma unroll
    for (int a = 0; a < 3; ++a) {
      float u = dx * sWp1[0 * 3 + a] + dy * sWp1[1 * 3 + a] +
                dz * sWp1[2 * 3 + a] + sbp1v[a];
      u = u * sps[a] + spt[a];
      tv[a] = fmaxf(u, 0.f);
    }
    float u0 = sbl1[0], u1 = sbl1[1];
#pragma unroll
    for (int c = 0; c < 16; ++c) {
      const float pe = tv[0] * sWp2[c] + tv[1] * sWp2[16 + c] +
                       tv[2] * sWp2[32 + c] + sbp2[c];
      float wc = xk[j * 16 + c] - q[c] + pe;
      wc = fmaxf(wc * s1[c] + t1[c], 0.f);
      u0 += wc * sWl1[c * 2 + 0];
      u1 += wc * sWl1[c * 2 + 1];
    }
    u0 = fmaxf(u0 * s2c[0] + t2c[0], 0.f);
    u1 = fmaxf(u1 * s2c[1] + t2c[1], 0.f);
    wvs[tid][n * 2 + 0] = u0 * sWl2[0] + u1 * sWl2[2] + sbl2[0];
    wvs[tid][n * 2 + 1] = u0 * sWl2[1] + u1 * sWl2[3] + sbl2[1];
  }

  // softmax over neighbors, per weight channel
  float m0 = -3.0e38f, m1 = -3.0e38f;
#pragma unroll 1
  for (int n = 0; n < NS; ++n) {
    m0 = fmaxf(m0, wvs[tid][n * 2 + 0]);
    m1 = fmaxf(m1, wvs[tid][n * 2 + 1]);
  }
  float sum0 = 0.f, sum1 = 0.f;
#pragma unroll 1
  for (int n = 0; n < NS; ++n) {
    const float e0 = __expf(wvs[tid][n * 2 + 0] - m0);
    const float e1 = __expf(wvs[tid][n * 2 + 1] - m1);
    wvs[tid][n * 2 + 0] = e0; wvs[tid][n * 2 + 1] = e1;
    sum0 += e0; sum1 += e1;
  }
  const float r0 = 1.f / sum0, r1 = 1.f / sum1;

  // pass 2: weighted aggregation (recompute p_e)
  float aggv[16];
#pragma unroll
  for (int c = 0; c < 16; ++c) aggv[c] = 0.f;
#pragma unroll 1
  for (int n = 0; n < NS; ++n) {
    const int j = nb[n];
    const float dx = xyz[j * 3 + 0] - qx;
    const float dy = xyz[j * 3 + 1] - qy;
    const float dz = xyz[j * 3 + 2] - qz;
    float tv[3];
#pragma unroll
    for (int a = 0; a < 3; ++a) {
      float u = dx * sWp1[0 * 3 + a] + dy * sWp1[1 * 3 + a] +
                dz * sWp1[2 * 3 + a] + sbp1v[a];
      u = u * sps[a] + spt[a];
      tv[a] = fmaxf(u, 0.f);
    }
    const float w0 = wvs[tid][n * 2 + 0] * r0;
    const float w1 = wvs[tid][n * 2 + 1] * r1;
#pragma unroll
    for (int c = 0; c < 16; ++c) {
      const float pe = tv[0] * sWp2[c] + tv[1] * sWp2[16 + c] +
                       tv[2] * sWp2[32 + c] + sbp2[c];
      // reshape (NS, SHARE=8, C_W=2): weight channel = c & 1
      aggv[c] += (xv[j * 16 + c] + pe) * ((c & 1) ? w1 : w0);
    }
  }
#pragma unroll
  for (int c = 0; c < 16; ++c) agg_out[i * 16 + c] = aggv[c];
}

// ---------------------------------------------------------------------------
// Kernel 4: out = agg @ W_eout + b_eout + x_fused  (K=16, 16 col-tiles x 4
// WMMAs). Fuses the residual read so x_fused is never materialized.
// ---------------------------------------------------------------------------
__global__ __launch_bounds__(256) void eout_kernel(
    const float* __restrict__ agg, const float* __restrict__ W_eout,
    const float* __restrict__ b_eout, const float* __restrict__ x_main,
    const float* __restrict__ x_mod, float* __restrict__ out) {
  __shared__ float sW[16 * 256];        // W_eout row-major copy

  const int tid  = threadIdx.x;
  const int wave = tid >> 5;
  const int lane = tid & 31;
  const int hi   = lane >> 4;
  const int lo   = lane & 15;
  const int rb   = (blockIdx.x * 8 + wave) * 16;

#pragma unroll
  for (int i = 0; i < 16; ++i) sW[tid * 16 + i] = W_eout[tid * 16 + i];
  __syncthreads();

  v2f af[4];
  const int arow = rb + lo;
#pragma unroll
  for (int t = 0; t < 4; ++t) {
    const int k0 = 4 * t + 2 * hi;
    af[t][0] = agg[arow * 16 + k0];
    af[t][1] = agg[arow * 16 + k0 + 1];
  }

  for (int ctile = 0; ctile < 16; ++ctile) {
    const int cc = ctile * 16 + lo;
    v8f acc = {0.f, 0.f, 0.f, 0.f, 0.f, 0.f, 0.f, 0.f};
#pragma unroll
    for (int t = 0; t < 4; ++t) {
      const int k0 = 4 * t + 2 * hi;
      v2f b; b[0] = sW[k0 * 256 + cc]; b[1] = sW[(k0 + 1) * 256 + cc];
      acc = wmma_f32_16x16x4(af[t], b, acc);
    }
    const float bias = b_eout[cc];
    const float* res = (cc < C_MAIN) ? (x_main + cc) : (x_mod + cc - C_MAIN);
#pragma unroll
    for (int r = 0; r < 8; ++r) {
      const int orow = rb + r + 8 * hi;
      out[orow * C_FUSED + cc] = acc[r] + bias + res[orow * C_MAIN];
    }
  }
}

// ---------------------------------------------------------------------------
extern "C" void kernel_launch(void* const* d_in, const int* in_sizes, int n_in,
                              void* d_out, int out_size, void* d_ws,
                              size_t ws_size, hipStream_t stream) {
  const float* x_main = (const float*)d_in[0];
  const float* x_mod  = (const float*)d_in[1];
  const float* xyz    = (const float*)d_in[2];
  const float* W_ein  = (const float*)d_in[3];
  const float* b_ein  = (const float*)d_in[4];
  const float* W_eout = (const float*)d_in[5];
  const float* b_eout = (const float*)d_in[6];
  const float* Wq = (const float*)d_in[7];
  const float* bq = (const float*)d_in[8];
  const float* Wk = (const float*)d_in[9];
  const float* bk = (const float*)d_in[10];
  const float* Wv = (const float*)d_in[11];
  const float* bv = (const float*)d_in[12];
  const float* Wp1 = (const float*)d_in[13];
  const float* bp1 = (const float*)d_in[14];
  const float* Wp2 = (const float*)d_in[15];
  const float* bp2 = (const float*)d_in[16];
  const float* Wl1 = (const float*)d_in[17];
  const float* bl1 = (const float*)d_in[18];
  const float* Wl2 = (const float*)d_in[19];
  const float* bl2 = (const float*)d_in[20];
  const float* bnp_g = (const float*)d_in[21];
  const float* bnp_b = (const float*)d_in[22];
  const float* bnp_m = (const float*)d_in[23];
  const float* bnp_v = (const float*)d_in[24];
  const float* bw1_g = (const float*)d_in[25];
  const float* bw1_b = (const float*)d_in[26];
  const float* bw1_m = (const float*)d_in[27];
  const float* bw1_v = (const float*)d_in[28];
  const float* bw2_g = (const float*)d_in[29];
  const float* bw2_b = (const float*)d_in[30];
  const float* bw2_m = (const float*)d_in[31];
  const float* bw2_v = (const float*)d_in[32];

  char* ws = (char*)d_ws;
  int*   idx = (int*)ws;                                   // N*16 ints (2 MB)
  float* xq  = (float*)(ws + (size_t)N * NS * sizeof(int));
  float* xk  = xq + (size_t)N * C_IN;
  float* xv  = xk + (size_t)N * C_IN;
  float* agg = xv + (size_t)N * C_IN;                      // total ~10 MB

  knn_kernel<<<N / 128, 256, 0, stream>>>(xyz, idx);
  ein_qkv_kernel<<<N / 128, 256, 0, stream>>>(x_main, x_mod, W_ein, b_ein, Wq,
                                              bq, Wk, bk, Wv, bv, xq, xk, xv);
  attn_kernel<<<N / 256, 256, 0, stream>>>(
      xyz, idx, xq, xk, xv, Wp1, bp1, Wp2, bp2, Wl1, bl1, Wl2, bl2, bnp_g,
      bnp_b, bnp_m, bnp_v, bw1_g, bw1_b, bw1_m, bw1_v, bw2_g, bw2_b, bw2_m,
      bw2_v, agg);
  eout_kernel<<<N / 128, 256, 0, stream>>>(agg, W_eout, b_eout, x_main, x_mod,
                                           (float*)d_out);
}